// DPSegmenter_41231686042033
// MI455X (gfx1250) — compile-verified
//
#include <hip/hip_runtime.h>
#include <hip/hip_bf16.h>

typedef __attribute__((ext_vector_type(2))) float v2f;
typedef __attribute__((ext_vector_type(8))) float v8f;

#define BB 16
#define TT 512
#define TROWS 513   // T+1
#define DD 256
#define KK 256
#define WIN 128     // max_seg_len
#define RING 132    // LDS ring rows: window [t-128,t-1] + row t + in-flight t+1

// ---------------------------------------------------------------------------
// Kernel 1: per-batch prefix sums.  P[b,t,:] = sum_{i<t} reps[b,i,:]
//           sq[b,t] = sum_{i<t} ||reps[b,i]||^2
// ---------------------------------------------------------------------------
__global__ void dpseg_prefix(const float* __restrict__ reps,
                             float* __restrict__ P,
                             float* __restrict__ sq) {
  const int b = blockIdx.x;
  const int d = threadIdx.x;
  __shared__ float red[256];
  float acc = 0.0f;
  float sacc = 0.0f;
  P[(b * TROWS + 0) * DD + d] = 0.0f;
  if (d == 0) sq[b * TROWS + 0] = 0.0f;
  for (int t = 1; t <= TT; ++t) {
    float x = reps[(b * TT + (t - 1)) * DD + d];
    acc += x;
    P[(b * TROWS + t) * DD + d] = acc;
    red[d] = x * x;
    __syncthreads();
    for (int off = 128; off > 0; off >>= 1) {
      if (d < off) red[d] += red[d + off];
      __syncthreads();
    }
    if (d == 0) {
      sacc += red[0];
      sq[b * TROWS + t] = sacc;
    }
    __syncthreads();
  }
}

// ---------------------------------------------------------------------------
// Kernel 2: cn[k] = ||codebook[k]||^2
// ---------------------------------------------------------------------------
__global__ void dpseg_cn(const float* __restrict__ cb, float* __restrict__ cn) {
  const int k = threadIdx.x;
  float s = 0.0f;
  for (int d = 0; d < DD; ++d) {
    float v = cb[k * DD + d];
    s += v * v;
  }
  cn[k] = s;
}

// ---------------------------------------------------------------------------
// Kernel 3: WMMA f32 GEMM  PC = P * codebook^T  fused with epilogue
//           F[m,k] = sq[m] - 2*PC[m,k] + t*cn[k]   (m = b*513 + t)
// grid = 513 blocks (16-row M tile), 128 threads = 4 waves, each wave a 16x64
// strip: 4 accumulators, 64 K-steps of V_WMMA_F32_16X16X4_F32.
// ---------------------------------------------------------------------------
__global__ void __launch_bounds__(128)
dpseg_gemmF(const float* __restrict__ P,
            const float* __restrict__ cb,
            const float* __restrict__ sq,
            const float* __restrict__ cn,
            float* __restrict__ F) {
  const int mt   = blockIdx.x;            // 0..512
  const int wave = threadIdx.x >> 5;      // 0..3
  const int lane = threadIdx.x & 31;
  const int m0 = mt * 16;
  const int n0 = wave * 64;
  const int r  = lane & 15;
  const int kh = lane >> 4;               // 0 or 1

  const float* Arow = P + (m0 + r) * DD;
  const float* B0 = cb + (n0 +  0 + r) * DD;
  const float* B1 = cb + (n0 + 16 + r) * DD;
  const float* B2 = cb + (n0 + 32 + r) * DD;
  const float* B3 = cb + (n0 + 48 + r) * DD;

  v8f acc0 = {}, acc1 = {}, acc2 = {}, acc3 = {};

  for (int kk = 0; kk < DD; kk += 4) {
    const int kp = kk + 2 * kh;
    v2f a  = *(const v2f*)(Arow + kp);
    v2f b0 = *(const v2f*)(B0 + kp);
    v2f b1 = *(const v2f*)(B1 + kp);
    v2f b2 = *(const v2f*)(B2 + kp);
    v2f b3 = *(const v2f*)(B3 + kp);
    acc0 = __builtin_amdgcn_wmma_f32_16x16x4_f32(false, a, false, b0, (short)0, acc0, false, false);
    acc1 = __builtin_amdgcn_wmma_f32_16x16x4_f32(false, a, false, b1, (short)0, acc1, false, false);
    acc2 = __builtin_amdgcn_wmma_f32_16x16x4_f32(false, a, false, b2, (short)0, acc2, false, false);
    acc3 = __builtin_amdgcn_wmma_f32_16x16x4_f32(false, a, false, b3, (short)0, acc3, false, false);
  }

  v8f accs[4] = {acc0, acc1, acc2, acc3};
#pragma unroll
  for (int j = 0; j < 4; ++j) {
    const int n = n0 + 16 * j + r;
    const float cnv = cn[n];
#pragma unroll
    for (int jj = 0; jj < 8; ++jj) {
      const int m = m0 + 8 * kh + jj;
      const int t = m % TROWS;
      F[m * KK + n] = sq[m] - 2.0f * accs[j][jj] + (float)t * cnv;
    }
  }
}

// ---------------------------------------------------------------------------
// Async global -> LDS helpers (CDNA5 GLOBAL_LOAD_ASYNC_TO_LDS, ASYNCcnt)
// One b128 moves 16B/lane; 32 lanes x 2 issues = one 1KB F row.
// ---------------------------------------------------------------------------
__device__ __forceinline__ void async_b128_to_lds(unsigned lds_byte,
                                                  unsigned gbyte_off,
                                                  const float* gbase) {
  asm volatile("global_load_async_to_lds_b128 %0, %1, %2 offset:0"
               :
               : "v"(lds_byte), "v"(gbyte_off), "s"(gbase)
               : "memory");
}

__device__ __forceinline__ void wait_asynccnt0() {
  asm volatile("s_wait_asynccnt 0x0" ::: "memory");
}

// ---------------------------------------------------------------------------
// Kernel 4: serial DP per batch.  One block per batch (16 blocks, 512 thr).
// 132-row LDS ring of F rows; row t+1 is DMA'd in asynchronously while step t
// computes, hidden behind ~0.5M VALU ops, then s_wait_asynccnt + barrier.
// seg_cost(s,t) = min_k ( F[t,k] - F[s,k] );  costs[t] = min_s costs[s]+seg.
// ---------------------------------------------------------------------------
__global__ void __launch_bounds__(512)
dpseg_dp(const float* __restrict__ F,
         float* __restrict__ costsG,
         int* __restrict__ prevG,
         int* __restrict__ tokG) {
  extern __shared__ float smem[];
  float* Fwin   = smem;                       // RING * 256
  float* costsL = Fwin + RING * KK;           // 513
  float* segmin = costsL + TROWS;             // 128
  int*   segtok = (int*)(segmin + WIN);       // 128

  const int b    = blockIdx.x;
  const int tid  = threadIdx.x;
  const int wave = tid >> 5;
  const int lane = tid & 31;
  const float* Fb = F + (size_t)b * TROWS * KK;
  const unsigned lds0 = __builtin_amdgcn_groupstaticsize();  // dynamic-LDS base

  if (tid == 0) {
    costsL[0] = 0.0f;
    prevG[b * TROWS + 0] = 0;
    tokG[b * TROWS + 0]  = 0;
  }

  // prologue: DMA rows 0 and 1 into ring slots 0,1
  if (wave == 0) {
#pragma unroll
    for (int r = 0; r < 2; ++r) {
#pragma unroll
      for (int c = 0; c < 2; ++c) {
        const unsigned part = (unsigned)(lane * 16 + c * 512);
        async_b128_to_lds(lds0 + (unsigned)r * 1024u + part,
                          (unsigned)r * 1024u + part, Fb);
      }
    }
    wait_asynccnt0();
  }
  __syncthreads();

  for (int t = 1; t <= TT; ++t) {
    // issue DMA of row t+1 into slot (t+1)%RING; that slot is outside the
    // live set [t-128, t] for this step, so no WAR hazard.
    if (wave == 0 && t < TT) {
      const unsigned slot = (unsigned)((t + 1) % RING);
#pragma unroll
      for (int c = 0; c < 2; ++c) {
        const unsigned part = (unsigned)(lane * 16 + c * 512);
        async_b128_to_lds(lds0 + slot * 1024u + part,
                          (unsigned)(t + 1) * 1024u + part, Fb);
      }
    }
    if (wave == 1 && lane == 0 && t + 2 <= TT) {
      __builtin_prefetch(Fb + (size_t)(t + 2) * KK, 0, 1);  // warm L2
    }

    // each lane caches its 8 F_t entries from the ring row t
    const float* Ftrow = Fwin + (t % RING) * KK;
    float ft[8];
#pragma unroll
    for (int c = 0; c < 8; ++c) ft[c] = Ftrow[lane + 32 * c];

    const int s_lo = (t > WIN) ? (t - WIN) : 0;

    // waves stride over valid s; 32 lanes x 8 chunks reduce over K=256
    for (int s = s_lo + wave; s < t; s += 16) {
      const float* rowp = Fwin + (s % RING) * KK;
      float bm = 3.4e38f;
      int   bi = 0;
#pragma unroll
      for (int c = 0; c < 8; ++c) {
        const int k = lane + 32 * c;
        const float v = ft[c] - rowp[k];
        if (v < bm) { bm = v; bi = k; }
      }
#pragma unroll
      for (int off = 16; off > 0; off >>= 1) {
        const float ov = __shfl_xor(bm, off, 32);
        const int   oi = __shfl_xor(bi, off, 32);
        if (ov < bm || (ov == bm && oi < bi)) { bm = ov; bi = oi; }
      }
      if (lane == 0) {
        segmin[s - s_lo] = bm;
        segtok[s - s_lo] = bi;
      }
    }
    __syncthreads();

    // wave 0 reduces over the window: argmin_s costs[s] + seg_cost
    if (wave == 0) {
      const int W = t - s_lo;
      float bv = 3.4e38f;
      int   bs = -1;
      int   btk = 0;
      for (int idx = lane; idx < W; idx += 32) {
        const int s = s_lo + idx;
        const float tot = costsL[s] + segmin[idx];
        if (tot < bv || (tot == bv && s < bs)) { bv = tot; bs = s; btk = segtok[idx]; }
      }
#pragma unroll
      for (int off = 16; off > 0; off >>= 1) {
        const float ov = __shfl_xor(bv, off, 32);
        const int   os = __shfl_xor(bs, off, 32);
        const int   ot = __shfl_xor(btk, off, 32);
        if (ov < bv || (ov == bv && os != -1 && (bs == -1 || os < bs))) {
          bv = ov; bs = os; btk = ot;
        }
      }
      if (lane == 0) {
        costsL[t] = bv;
        prevG[b * TROWS + t] = bs;
        tokG[b * TROWS + t]  = btk;
      }
      // make the in-flight DMA of row t+1 visible before next step
      wait_asynccnt0();
    }
    __syncthreads();
  }

  if (tid == 0) costsG[b] = costsL[TT];
}

// ---------------------------------------------------------------------------
// Kernel 5: backtrace + pack outputs (boundaries, tokens, valid, final_cost)
// ---------------------------------------------------------------------------
__global__ void dpseg_backtrace(const int* __restrict__ prevG,
                                const int* __restrict__ tokG,
                                const float* __restrict__ costsG,
                                float* __restrict__ out) {
  const int b = threadIdx.x;  // 16 threads
  if (b >= BB) return;
  int st = TT;
  for (int i = 0; i < TT; ++i) {
    out[0 * BB * TT + b * TT + i] = (float)st;
    out[1 * BB * TT + b * TT + i] = (float)tokG[b * TROWS + st];
    out[2 * BB * TT + b * TT + i] = (st != 0) ? 1.0f : 0.0f;
    st = (st != 0) ? prevG[b * TROWS + st] : 0;
  }
  out[3 * BB * TT + b] = costsG[b];
}

// ---------------------------------------------------------------------------
extern "C" void kernel_launch(void* const* d_in, const int* in_sizes, int n_in,
                              void* d_out, int out_size, void* d_ws, size_t ws_size,
                              hipStream_t stream) {
  const float* reps = (const float*)d_in[0];   // B*T*D f32
  const float* cb   = (const float*)d_in[1];   // K*D f32
  // d_in[2] = max_seg_len (=128, compiled in as WIN)

  float* ws = (float*)d_ws;
  float* P      = ws;                                   // BB*TROWS*DD
  float* Ff     = P + (size_t)BB * TROWS * DD;          // BB*TROWS*KK
  float* sq     = Ff + (size_t)BB * TROWS * KK;         // BB*TROWS
  float* cn     = sq + BB * TROWS;                      // KK
  float* costsG = cn + KK;                              // BB
  int*   prevG  = (int*)(costsG + BB);                  // BB*TROWS
  int*   tokG   = prevG + BB * TROWS;                   // BB*TROWS

  dpseg_prefix<<<BB, 256, 0, stream>>>(reps, P, sq);
  dpseg_cn<<<1, 256, 0, stream>>>(cb, cn);
  dpseg_gemmF<<<TROWS, 128, 0, stream>>>(P, cb, sq, cn, Ff);

  const size_t dp_smem =
      (size_t)(RING * KK + TROWS + WIN) * sizeof(float) + WIN * sizeof(int);
  dpseg_dp<<<BB, 512, dp_smem, stream>>>(Ff, costsG, prevG, tokG);

  dpseg_backtrace<<<1, 32, 0, stream>>>(prevG, tokG, costsG, (float*)d_out);
}